// SpatialAttentionLayer_1580547973359
// MI455X (gfx1250) — compile-verified
//
#include <hip/hip_runtime.h>
#include <math.h>

// Problem constants from the reference
#define B_  32
#define C_  256
#define S_  64
#define HW  (S_ * S_)            // 4096
#define N_ELEM ((size_t)B_ * C_ * HW)   // 33,554,432 floats (128 MB)

typedef __attribute__((ext_vector_type(2))) float v2f;
typedef __attribute__((ext_vector_type(4))) float v4f;
typedef __attribute__((ext_vector_type(8))) float v8f;

// ---------------------------------------------------------------------------
// Kernel 1: per-(b,h) channel reduction: max & mean over C for every w.
// Block = 256 threads: 16 w-groups (float4 each) x 16 channel groups.
// Regular-temporal loads on purpose: x (128 MB) fits in the 192 MB L2, and
// pass 4 re-reads it — let these loads warm L2.
// ---------------------------------------------------------------------------
__global__ void channel_reduce(const float* __restrict__ x,
                               float* __restrict__ spMax,
                               float* __restrict__ spMean) {
    const int h   = blockIdx.x;          // 0..63
    const int b   = blockIdx.y;          // 0..31
    const int tid = threadIdx.x;         // 0..255
    const int w4  = tid & 15;            // which float4 of the 64-wide row
    const int cg  = tid >> 4;            // channel group 0..15

    const float* base = x + (((size_t)b * C_) * S_ + h) * S_ + w4 * 4;

    float4 mx = make_float4(-3.4028235e38f, -3.4028235e38f,
                            -3.4028235e38f, -3.4028235e38f);
    float4 sm = make_float4(0.f, 0.f, 0.f, 0.f);

    #pragma unroll 4
    for (int c = cg; c < C_; c += 16) {
        const float4 v = *(const float4*)(base + (size_t)c * HW);
        sm.x += v.x; sm.y += v.y; sm.z += v.z; sm.w += v.w;
        mx.x = fmaxf(mx.x, v.x); mx.y = fmaxf(mx.y, v.y);
        mx.z = fmaxf(mx.z, v.z); mx.w = fmaxf(mx.w, v.w);
    }

    __shared__ float4 sMax[256];
    __shared__ float4 sSum[256];
    sMax[tid] = mx;
    sSum[tid] = sm;
    __syncthreads();

    // reduce over channel groups (stride multiples of 16 keep w4 fixed)
    for (int s = 128; s >= 16; s >>= 1) {
        if (tid < s) {
            float4 a = sMax[tid], c2 = sMax[tid + s];
            a.x = fmaxf(a.x, c2.x); a.y = fmaxf(a.y, c2.y);
            a.z = fmaxf(a.z, c2.z); a.w = fmaxf(a.w, c2.w);
            sMax[tid] = a;
            float4 u = sSum[tid], v = sSum[tid + s];
            u.x += v.x; u.y += v.y; u.z += v.z; u.w += v.w;
            sSum[tid] = u;
        }
        __syncthreads();
    }

    if (tid < 16) {
        const float inv = 1.0f / (float)C_;
        float4 m = sMax[tid];
        float4 u = sSum[tid];
        u.x *= inv; u.y *= inv; u.z *= inv; u.w *= inv;
        const size_t o = (size_t)b * HW + h * S_ + tid * 4;
        *(float4*)(spMax  + o) = m;
        *(float4*)(spMean + o) = u;
    }
}

// ---------------------------------------------------------------------------
// Kernel 2: per-batch width/height stats + both "branch" computations.
// 256 threads = 4 groups of 64; groups 0/1 compute w/h branches, groups 2/3
// duplicate them (results discarded) so every wave hits every barrier.
// ---------------------------------------------------------------------------
__device__ __forceinline__ float grp_reduce_sum(float* r, int j) {
    __syncthreads();
    for (int s = 32; s > 0; s >>= 1) {
        if (j < s) r[j] += r[j + s];
        __syncthreads();
    }
    float v = r[0];
    __syncthreads();
    return v;
}

__device__ __forceinline__ float grp_reduce_max(float* r, int j) {
    __syncthreads();
    for (int s = 32; s > 0; s >>= 1) {
        if (j < s) r[j] = fmaxf(r[j], r[j + s]);
        __syncthreads();
    }
    float v = r[0];
    __syncthreads();
    return v;
}

__global__ void stats_branch(const float* __restrict__ spMax,
                             const float* __restrict__ spMean,
                             const float* __restrict__ ww0, const float* __restrict__ ww1,
                             const float* __restrict__ ww2, const float* __restrict__ wb0,
                             const float* __restrict__ wb1, const float* __restrict__ wb2,
                             const float* __restrict__ hw0, const float* __restrict__ hw1,
                             const float* __restrict__ hw2, const float* __restrict__ hb0,
                             const float* __restrict__ hb1, const float* __restrict__ hb2,
                             float* __restrict__ yW, float* __restrict__ yH) {
    const int b   = blockIdx.x;
    const int tid = threadIdx.x;

    __shared__ float smax[S_ * (S_ + 1)];   // padded rows: no bank conflicts
    __shared__ float smean[S_ * (S_ + 1)];
    __shared__ float wMean[S_], wMax[S_], hMeanA[S_], hMaxA[S_];
    __shared__ float red[4][S_];

    for (int i = tid; i < HW; i += 256) {
        const int h = i >> 6, w = i & 63;
        smax [h * (S_ + 1) + w] = spMax [(size_t)b * HW + i];
        smean[h * (S_ + 1) + w] = spMean[(size_t)b * HW + i];
    }
    __syncthreads();

    if (tid < 64) {                 // width stats: reduce over h
        const int w = tid;
        float mx = -3.4028235e38f, sm = 0.f;
        for (int h = 0; h < S_; ++h) {
            mx = fmaxf(mx, smax[h * (S_ + 1) + w]);
            sm += smean[h * (S_ + 1) + w];
        }
        wMax[w] = mx; wMean[w] = sm * (1.0f / S_);
    } else if (tid < 128) {         // height stats: reduce over w
        const int h = tid - 64;
        float mx = -3.4028235e38f, sm = 0.f;
        for (int w = 0; w < S_; ++w) {
            mx = fmaxf(mx, smax[h * (S_ + 1) + w]);
            sm += smean[h * (S_ + 1) + w];
        }
        hMaxA[h] = mx; hMeanA[h] = sm * (1.0f / S_);
    }
    __syncthreads();

    const int g  = tid >> 6;        // 0..3
    const int j  = tid & 63;
    const int br = g & 1;           // 0 = width branch, 1 = height branch

    const float mv  = br ? hMeanA[j] : wMean[j];
    const float mxv = br ? hMaxA[j]  : wMax[j];
    const float* w0 = br ? hw0 : ww0;
    const float* w1 = br ? hw1 : ww1;
    const float* w2 = br ? hw2 : ww2;
    const float* b0 = br ? hb0 : wb0;
    const float* b1 = br ? hb1 : wb1;
    const float* b2 = br ? hb2 : wb2;
    float* r = red[g];

    // softmax(mean_vec)
    r[j] = mv;
    const float m = grp_reduce_max(r, j);
    const float e = expf(mv - m);
    r[j] = e;
    const float denom = grp_reduce_sum(r, j);
    const float sfx = e / denom;

    // y0 = mean_vec . w0 ; y1 = max_vec . w1
    r[j] = mv * w0[j];
    const float y0 = grp_reduce_sum(r, j);
    r[j] = mxv * w1[j];
    const float y1 = grp_reduce_sum(r, j);

    const float y0t = tanhf(y0 * sfx + b0[j]);
    const float y1t = tanhf(y1 * sfx + b1[j]);

    r[j] = y1t * w2[j];
    const float tsum = grp_reduce_sum(r, j);

    const float out = tsum * y0t + b2[j];
    if (g == 0) yW[b * S_ + j] = out;
    if (g == 1) yH[b * S_ + j] = out;
}

// ---------------------------------------------------------------------------
// Kernel 3: spatial = tanh(y_h (x) y_w) + 1, via V_WMMA_F32_16X16X4_F32.
// Outer product == GEMM with K=1 (pad K to 4 with zeros). Exact fp32 math.
// One wave (32 lanes) per batch; 16 WMMAs cover the 64x64 map.
// A layout: lanes 0-15 carry M rows (VGPR0=K0), lanes 16-31 carry K2/K3 -> 0.
// ---------------------------------------------------------------------------
__global__ __launch_bounds__(32) void spatial_outer_wmma(
        const float* __restrict__ yH, const float* __restrict__ yW,
        float* __restrict__ spat1) {
    const int b    = blockIdx.x;
    const int lane = threadIdx.x;               // 0..31, EXEC all-ones
    const float* yh = yH + b * S_;
    const float* yw = yW + b * S_;
    const int lo = lane & 15;
    const bool low = lane < 16;

    #pragma unroll
    for (int mt = 0; mt < 4; ++mt) {
        v2f a;
        a.x = low ? yh[mt * 16 + lo] : 0.0f;    // K=0 column only
        a.y = 0.0f;                              // K=1 (or K=3) zero
        #pragma unroll
        for (int nt = 0; nt < 4; ++nt) {
            v2f bv;
            bv.x = low ? yw[nt * 16 + lo] : 0.0f; // K=0 row only
            bv.y = 0.0f;
            v8f c = {};
            v8f d = __builtin_amdgcn_wmma_f32_16x16x4_f32(
                        false, a, false, bv, (short)0, c, false, false);
            const int mbase = mt * 16 + (low ? 0 : 8);
            const int n     = nt * 16 + lo;
            float* dst = spat1 + (size_t)b * HW + n;
            #pragma unroll
            for (int r8 = 0; r8 < 8; ++r8)
                dst[(mbase + r8) * S_] = tanhf(d[r8]) + 1.0f;
        }
    }
}

// ---------------------------------------------------------------------------
// Kernel 4: out = x * (spatial + 1). Pure float4 streaming.
// x reads: non-temporal (last use; hit lines left in L2 by pass 1, then drop).
// out stores: non-temporal so 128 MB of output doesn't evict x from L2
// before its lines are re-read (x fits entirely in the 192 MB L2).
// spat1 loads: regular temporal (tiny, highly reused).
// ---------------------------------------------------------------------------
__global__ void scale_out(const float* __restrict__ x,
                          const float* __restrict__ spat1,
                          float* __restrict__ out) {
    const size_t i = (size_t)blockIdx.x * blockDim.x + threadIdx.x; // float4 idx
    const v4f xv = __builtin_nontemporal_load(((const v4f*)x) + i);
    const size_t b   = i >> 18;          // / (C*S*S/4) = 262144
    const size_t hw4 = i & 1023;         // float4 within the 64x64 map
    const v4f sv = ((const v4f*)spat1)[b * (HW / 4) + hw4];
    const v4f o = xv * sv;
    __builtin_nontemporal_store(o, ((v4f*)out) + i);
}

// ---------------------------------------------------------------------------
extern "C" void kernel_launch(void* const* d_in, const int* in_sizes, int n_in,
                              void* d_out, int out_size, void* d_ws, size_t ws_size,
                              hipStream_t stream) {
    const float* x   = (const float*)d_in[0];
    const float* ww0 = (const float*)d_in[1];
    const float* ww1 = (const float*)d_in[2];
    const float* ww2 = (const float*)d_in[3];
    const float* wb0 = (const float*)d_in[4];
    const float* wb1 = (const float*)d_in[5];
    const float* wb2 = (const float*)d_in[6];
    const float* hw0 = (const float*)d_in[7];
    const float* hw1 = (const float*)d_in[8];
    const float* hw2 = (const float*)d_in[9];
    const float* hb0 = (const float*)d_in[10];
    const float* hb1 = (const float*)d_in[11];
    const float* hb2 = (const float*)d_in[12];

    // Workspace layout (floats); ~1.55 MB total, all float4-aligned.
    float* ws     = (float*)d_ws;
    float* spMax  = ws;                       // 32*64*64 = 131072
    float* spMean = ws + 131072;              // 131072
    float* yW     = ws + 262144;              // 2048
    float* yH     = ws + 264192;              // 2048
    float* spat1  = ws + 266240;              // 131072

    channel_reduce<<<dim3(S_, B_), 256, 0, stream>>>(x, spMax, spMean);
    stats_branch<<<B_, 256, 0, stream>>>(spMax, spMean,
                                         ww0, ww1, ww2, wb0, wb1, wb2,
                                         hw0, hw1, hw2, hb0, hb1, hb2,
                                         yW, yH);
    spatial_outer_wmma<<<B_, 32, 0, stream>>>(yH, yW, spat1);
    scale_out<<<(unsigned)(N_ELEM / 4 / 256), 256, 0, stream>>>(x, spat1, (float*)d_out);
}